// ROILayer_72490458022384
// MI455X (gfx1250) — compile-verified
//
#include <hip/hip_runtime.h>
#include <hip/hip_bf16.h>

// ---------------------------------------------------------------------------
// Problem constants (from reference)
// ---------------------------------------------------------------------------
#define BATCH      64
#define NCAND      10647
#define ROWLEN     85          // 4 box + 1 conf + 80 cls
#define NCLASS     80
#define TOPK       512
#define NTILES     4
#define TILE_SZ    104.0f      // 416 / 4
#define CONF_THR   0.3f
#define NMS_THR    0.2f
#define FEATW      (NTILES * NCLASS)   // 320

#define CHUNK      2048
#define NCHUNK     6           // 6*2048 = 12288 >= 10647
#define SORT2      4096        // merge-sort size (6*512 padded)

typedef __attribute__((ext_vector_type(2))) float v2f;
typedef __attribute__((ext_vector_type(8))) float v8f;

// ---------------------------------------------------------------------------
// Bitonic sort (descending) of u64 keys in LDS. nthreads uniform across block.
// ---------------------------------------------------------------------------
__device__ inline void bitonic_desc(unsigned long long* a, int n, int tid, int nthreads) {
    for (int k = 2; k <= n; k <<= 1) {
        for (int j = k >> 1; j > 0; j >>= 1) {
            for (int t = tid; t < n; t += nthreads) {
                int ixj = t ^ j;
                if (ixj > t) {
                    bool big_first = ((t & k) == 0);
                    unsigned long long A = a[t], B = a[ixj];
                    bool swp = big_first ? (A < B) : (A > B);
                    if (swp) { a[t] = B; a[ixj] = A; }
                }
            }
            __syncthreads();
        }
    }
}

// ---------------------------------------------------------------------------
// Kernel 1a: per-(batch,chunk) scoring + chunk-local top-512 selection.
// grid = BATCH*NCHUNK blocks, 512 threads (16 waves).  This is the phase that
// streams 232 MB, so it gets 6x the workgroups of the old fused version.
// ---------------------------------------------------------------------------
__global__ __launch_bounds__(512)
void k1a_score_chunk(const float* __restrict__ x,
                     unsigned long long* __restrict__ keysW) {
    __shared__ unsigned long long ck[CHUNK];     // 16 KB only -> high residency

    const int blk   = blockIdx.x;
    const int b     = blk / NCHUNK;
    const int chunk = blk % NCHUNK;
    const int tid   = threadIdx.x;
    const int wave  = tid >> 5;
    const int lane  = tid & 31;
    const float* xb = x + (size_t)b * NCAND * ROWLEN;

    // one wave per candidate: coalesced class reads + wave max-reduce (wave32)
    for (int local = wave; local < CHUNK; local += 16) {
        const int n = chunk * CHUNK + local;
        unsigned long long key = 0ull;   // padding sorts last
        if (n < NCAND) {
            const float* row = xb + (size_t)n * ROWLEN;
            __builtin_prefetch(row + 16 * ROWLEN, 0, 1);  // global_prefetch_b8
            float m = -3.4e38f;
            if (lane      < NCLASS) m = fmaxf(m, row[5 + lane]);
            if (lane + 32 < NCLASS) m = fmaxf(m, row[5 + lane + 32]);
            if (lane + 64 < NCLASS) m = fmaxf(m, row[5 + lane + 64]);
            #pragma unroll
            for (int off = 16; off > 0; off >>= 1)
                m = fmaxf(m, __shfl_xor(m, off, 32));
            const float conf  = row[4];
            const float score = (conf >= CONF_THR) ? conf * m : -1.0e9f;
            unsigned int s = __float_as_uint(score);
            s ^= (s >> 31) ? 0xFFFFFFFFu : 0x80000000u;     // monotonic map
            key = ((unsigned long long)s << 32) |
                  (unsigned long long)(0xFFFFFFFFu - (unsigned int)n);
        }
        if (lane == 0) ck[local] = key;
    }
    __syncthreads();
    bitonic_desc(ck, CHUNK, tid, 512);
    // spill chunk's top-512 keys
    keysW[(size_t)blk * TOPK + tid] = ck[tid];
}

// ---------------------------------------------------------------------------
// Kernel 1b: per-batch merge of the 6x512 survivors, final sorted top-512,
// gather box/conf/cls/valid.  grid = BATCH blocks, 512 threads.
// ---------------------------------------------------------------------------
__global__ __launch_bounds__(512)
void k1b_merge_gather(const float* __restrict__ x,
                      const unsigned long long* __restrict__ keysW,
                      float* __restrict__ boxesG, float* __restrict__ confG,
                      int* __restrict__ clsG, int* __restrict__ validG) {
    __shared__ unsigned long long stage[SORT2];  // 32 KB

    const int b   = blockIdx.x;
    const int tid = threadIdx.x;
    const float* xb = x + (size_t)b * NCAND * ROWLEN;
    const unsigned long long* kb = keysW + (size_t)b * (NCHUNK * TOPK);

    #pragma unroll
    for (int c = 0; c < NCHUNK; ++c)
        stage[c * TOPK + tid] = kb[c * TOPK + tid];
    stage[NCHUNK * TOPK + tid]       = 0ull;     // pad 3072..4095
    stage[NCHUNK * TOPK + 512 + tid] = 0ull;
    __syncthreads();
    bitonic_desc(stage, SORT2, tid, 512);

    // gather: thread t owns rank t
    unsigned long long key = stage[tid];
    unsigned int s = (unsigned int)(key >> 32);
    s ^= (s >> 31) ? 0x80000000u : 0xFFFFFFFFu;   // inverse monotonic map
    const float score = __uint_as_float(s);
    int idx   = (int)(0xFFFFFFFFu - (unsigned int)(key & 0xFFFFFFFFu));
    int valid = (score > -1.0e8f) ? 1 : 0;
    if (idx < 0 || idx >= NCAND) { idx = 0; valid = 0; }

    const float* row = xb + (size_t)idx * ROWLEN;
    const float cx = row[0], cy = row[1], w = row[2], h = row[3];
    const float conf = row[4];
    float best = row[5]; int bi = 0;
    for (int c = 1; c < NCLASS; ++c) {
        float v = row[5 + c];
        if (v > best) { best = v; bi = c; }
    }
    const size_t base = (size_t)b * TOPK + tid;
    boxesG[base * 4 + 0] = cx - 0.5f * w;
    boxesG[base * 4 + 1] = cy - 0.5f * h;
    boxesG[base * 4 + 2] = cx + 0.5f * w;
    boxesG[base * 4 + 3] = cy + 0.5f * h;
    confG[base]  = conf;
    clsG[base]   = bi;
    validG[base] = valid;
}

// ---------------------------------------------------------------------------
// Kernel 2: sequential merge-NMS (512 iterations) + tile/class scatter.
// One block per batch, 512 threads.  Initial state load uses the CDNA5
// async global->LDS path (GLOBAL_LOAD_ASYNC_TO_LDS_B32, ASYNCcnt).
// ---------------------------------------------------------------------------
__device__ inline int tile_idx(float v) {
    int t = (int)floorf(v * (1.0f / TILE_SZ));
    return min(max(t, 0), NTILES - 1);
}

__device__ inline void async_g2l_b32(void* ldsDst, const void* gSrc) {
    unsigned lds_off = (unsigned)(unsigned long long)(uintptr_t)ldsDst; // addr[31:0]
    asm volatile("global_load_async_to_lds_b32 %0, %1, off"
                 :: "v"(lds_off), "v"(gSrc) : "memory");
}

__global__ __launch_bounds__(512)
void k2_nms_feats(const float* __restrict__ boxesG, const float* __restrict__ confG,
                  const int* __restrict__ clsG, const int* __restrict__ validG,
                  float* __restrict__ featx, float* __restrict__ featy) {
    __shared__ float bx0[TOPK], bx1[TOPK], bx2[TOPK], bx3[TOPK];
    __shared__ float cf[TOPK];
    __shared__ int   cl[TOPK];
    __shared__ int   alive[TOPK];
    __shared__ int   keptS[TOPK];
    __shared__ float red[5];
    __shared__ float fx[FEATW], fy[FEATW];

    const int b = blockIdx.x, tid = threadIdx.x, lane = tid & 31;
    const size_t base = (size_t)b * TOPK + tid;

    // async copy global -> LDS (7 dwords per lane), then wait + barrier
    async_g2l_b32(&bx0[tid], boxesG + base * 4 + 0);
    async_g2l_b32(&bx1[tid], boxesG + base * 4 + 1);
    async_g2l_b32(&bx2[tid], boxesG + base * 4 + 2);
    async_g2l_b32(&bx3[tid], boxesG + base * 4 + 3);
    async_g2l_b32(&cf[tid],  confG  + base);
    async_g2l_b32(&cl[tid],  clsG   + base);
    async_g2l_b32(&alive[tid], validG + base);
    keptS[tid] = 0;
    if (tid < FEATW) { fx[tid] = 0.0f; fy[tid] = 0.0f; }
    asm volatile("s_wait_asynccnt 0x0" ::: "memory");
    __syncthreads();

    for (int i = 0; i < TOPK; ++i) {
        if (tid < 5) red[tid] = 0.0f;
        __syncthreads();

        const int   keep_i = alive[i];
        const float ib0 = bx0[i], ib1 = bx1[i], ib2 = bx2[i], ib3 = bx3[i];
        const int   ic  = cl[i];

        const float jb0 = bx0[tid], jb1 = bx1[tid], jb2 = bx2[tid], jb3 = bx3[tid];
        const float ix1 = fmaxf(ib0, jb0), iy1 = fmaxf(ib1, jb1);
        const float ix2 = fminf(ib2, jb2), iy2 = fminf(ib3, jb3);
        const float inter = fmaxf(ix2 - ix1 + 1.0f, 0.0f) * fmaxf(iy2 - iy1 + 1.0f, 0.0f);
        const float a1 = (ib2 - ib0 + 1.0f) * (ib3 - ib1 + 1.0f);
        const float a2 = (jb2 - jb0 + 1.0f) * (jb3 - jb1 + 1.0f);
        const float iou = inter / (a1 + a2 - inter + 1e-16f);
        const int inv = (iou > NMS_THR) && (cl[tid] == ic) && alive[tid];
        const float w = inv ? cf[tid] : 0.0f;

        float s0 = w, s1 = w * jb0, s2 = w * jb1, s3 = w * jb2, s4 = w * jb3;
        #pragma unroll
        for (int off = 16; off > 0; off >>= 1) {
            s0 += __shfl_xor(s0, off, 32);
            s1 += __shfl_xor(s1, off, 32);
            s2 += __shfl_xor(s2, off, 32);
            s3 += __shfl_xor(s3, off, 32);
            s4 += __shfl_xor(s4, off, 32);
        }
        if (lane == 0) {
            atomicAdd(&red[0], s0); atomicAdd(&red[1], s1);
            atomicAdd(&red[2], s2); atomicAdd(&red[3], s3);
            atomicAdd(&red[4], s4);
        }
        __syncthreads();

        if (keep_i) {
            if (tid == 0) {
                const float rw = 1.0f / (red[0] + 1e-16f);
                bx0[i] = red[1] * rw; bx1[i] = red[2] * rw;
                bx2[i] = red[3] * rw; bx3[i] = red[4] * rw;
                keptS[i] = 1;
            }
            if (inv) alive[tid] = 0;   // includes i itself (iou == 1)
        }
        __syncthreads();
    }

    // tile/class scatter (reference adds w=0 for non-kept -> skip them)
    if (keptS[tid]) {
        const float w  = cf[tid];
        const int   c  = cl[tid];
        const int   xt  = tile_idx(bx0[tid]);
        const int   yt  = tile_idx(bx1[tid]);
        const int   xt2 = tile_idx(bx3[tid]);
        const int   yt2 = tile_idx(cf[tid]);
        atomicAdd(&fx[xt * NCLASS + c], w);
        if (xt2 != xt) atomicAdd(&fx[xt2 * NCLASS + c], w);
        atomicAdd(&fy[yt * NCLASS + c], w);
        if (yt2 != yt) atomicAdd(&fy[yt2 * NCLASS + c], w);
    }
    __syncthreads();
    if (tid < FEATW) {
        featx[(size_t)b * FEATW + tid] = fx[tid];
        featy[(size_t)b * FEATW + tid] = fy[tid];
    }
}

// ---------------------------------------------------------------------------
// Kernel 3: two 4-layer MLPs via V_WMMA_F32_16X16X4_F32.
// grid = 2 blocks (head x / head y), 512 threads = 16 waves, 1 wave per tile.
// ---------------------------------------------------------------------------
struct MlpParams {
    const float *W1, *b1, *W2, *b2, *W3, *b3, *W4, *b4;
};

// D(16x16) += A(16x4) x B(4x16) over K, f32 WMMA. nValid guards ragged N.
__device__ inline v8f wmma_gemm_tile(const float* __restrict__ A, int lda,
                                     const float* __restrict__ B, int ldb,
                                     int K, int mBase, int nBase, int nValid,
                                     int lane) {
    const int half = lane >> 4;        // K sub-pair selector per ISA layout
    const int l15  = lane & 15;
    const int m    = mBase + l15;
    const int n    = nBase + l15;
    const bool nOk = (l15 < nValid);
    v8f c = {};
    for (int kb = 0; kb < K; kb += 4) {
        const int ka = kb + 2 * half;
        v2f a, bb;
        a.x  = A[(size_t)m * lda + ka];
        a.y  = A[(size_t)m * lda + ka + 1];
        bb.x = nOk ? B[(size_t)(ka)     * ldb + n] : 0.0f;
        bb.y = nOk ? B[(size_t)(ka + 1) * ldb + n] : 0.0f;
        c = __builtin_amdgcn_wmma_f32_16x16x4_f32(
                /*neg_a=*/false, a, /*neg_b=*/false, bb,
                /*c_mod=*/(short)0, c, /*reuse_a=*/false, /*reuse_b=*/false);
    }
    return c;
}

__device__ inline void store_act(float* H, int ldh, v8f c, int mBase, int nBase,
                                 int nValid, const float* bias, bool doRelu, int lane) {
    const int half = lane >> 4;
    const int l15  = lane & 15;
    if (l15 < nValid) {
        const float bv = bias[nBase + l15];
        #pragma unroll
        for (int r = 0; r < 8; ++r) {
            float v = c[r] + bv;
            if (doRelu) v = fmaxf(v, 0.0f);
            H[(size_t)(mBase + r + 8 * half) * ldh + (nBase + l15)] = v;
        }
    }
}

__global__ __launch_bounds__(512)
void k3_mlp(const float* __restrict__ featx, const float* __restrict__ featy,
            MlpParams px, MlpParams py, float* __restrict__ out) {
    __shared__ float h1[64 * 64];   // 16 KB
    __shared__ float h2[64 * 32];   //  8 KB
    __shared__ float h3[64 * 24];   //  6 KB

    const int head = blockIdx.x;
    const int tid  = threadIdx.x;
    const int wave = tid >> 5;
    const int lane = tid & 31;

    const float* feat = head ? featy : featx;
    const MlpParams P = head ? py : px;
    float* outH = out + head * (BATCH * NTILES);

    // L1: (64x320) x (320x64) -> h1, relu. 16 tiles, 1 per wave.
    {
        const int mt = wave >> 2, nt = wave & 3;
        v8f c = wmma_gemm_tile(feat, FEATW, P.W1, 64, FEATW, mt * 16, nt * 16, 16, lane);
        store_act(h1, 64, c, mt * 16, nt * 16, 16, P.b1, true, lane);
    }
    __syncthreads();

    // L2: (64x64) x (64x32) -> h2, relu. 8 tiles (waves 0..7, full EXEC inside).
    if (wave < 8) {
        const int mt = wave >> 1, nt = wave & 1;
        v8f c = wmma_gemm_tile(h1, 64, P.W2, 32, 64, mt * 16, nt * 16, 16, lane);
        store_act(h2, 32, c, mt * 16, nt * 16, 16, P.b2, true, lane);
    }
    __syncthreads();

    // L3: (64x32) x (32x24) -> h3, relu. 8 tiles; second N-tile has 8 valid cols.
    if (wave < 8) {
        const int mt = wave >> 1, nt = wave & 1;
        const int nValid = nt ? 8 : 16;
        v8f c = wmma_gemm_tile(h2, 32, P.W3, 24, 32, mt * 16, nt * 16, nValid, lane);
        store_act(h3, 24, c, mt * 16, nt * 16, nValid, P.b3, true, lane);
    }
    __syncthreads();

    // L4: (64x24) x (24x4) -> out, no relu. 4 tiles, nValid = 4.
    if (wave < 4) {
        const int mt = wave;
        v8f c = wmma_gemm_tile(h3, 24, P.W4, 4, 24, mt * 16, 0, 4, lane);
        const int half = lane >> 4, l15 = lane & 15;
        if (l15 < 4) {
            const float bv = P.b4[l15];
            #pragma unroll
            for (int r = 0; r < 8; ++r) {
                const int m = mt * 16 + r + 8 * half;
                outH[m * NTILES + l15] = c[r] + bv;
            }
        }
    }
}

// ---------------------------------------------------------------------------
// Launch
// ---------------------------------------------------------------------------
extern "C" void kernel_launch(void* const* d_in, const int* in_sizes, int n_in,
                              void* d_out, int out_size, void* d_ws, size_t ws_size,
                              hipStream_t stream) {
    const float* x = (const float*)d_in[0];

    MlpParams px, py;
    px.W1 = (const float*)d_in[1];  px.b1 = (const float*)d_in[2];
    px.W2 = (const float*)d_in[3];  px.b2 = (const float*)d_in[4];
    px.W3 = (const float*)d_in[5];  px.b3 = (const float*)d_in[6];
    px.W4 = (const float*)d_in[7];  px.b4 = (const float*)d_in[8];
    py.W1 = (const float*)d_in[9];  py.b1 = (const float*)d_in[10];
    py.W2 = (const float*)d_in[11]; py.b2 = (const float*)d_in[12];
    py.W3 = (const float*)d_in[13]; py.b3 = (const float*)d_in[14];
    py.W4 = (const float*)d_in[15]; py.b4 = (const float*)d_in[16];

    // workspace layout (u64 keys first for alignment)
    unsigned long long* keysW = (unsigned long long*)d_ws;       // 64*6*512 u64 = 1.5 MB
    float* boxes  = (float*)(keysW + (size_t)BATCH * NCHUNK * TOPK); // 64*512*4
    float* confA  = boxes + (size_t)BATCH * TOPK * 4;            // 64*512
    int*   clsA   = (int*)(confA + (size_t)BATCH * TOPK);        // 64*512
    int*   validA = clsA + (size_t)BATCH * TOPK;                 // 64*512
    float* featx  = (float*)(validA + (size_t)BATCH * TOPK);     // 64*320
    float* featy  = featx + (size_t)BATCH * FEATW;               // 64*320

    k1a_score_chunk <<<BATCH * NCHUNK, 512, 0, stream>>>(x, keysW);
    k1b_merge_gather<<<BATCH,          512, 0, stream>>>(x, keysW, boxes, confA, clsA, validA);
    k2_nms_feats    <<<BATCH,          512, 0, stream>>>(boxes, confA, clsA, validA, featx, featy);
    k3_mlp          <<<2,              512, 0, stream>>>(featx, featy, px, py, (float*)d_out);
}